// DualSDMCC_36636071035177
// MI455X (gfx1250) — compile-verified
//
#include <hip/hip_runtime.h>

// ---------------------------------------------------------------------------
// DualSDMCC on MI455X (gfx1250): fp32 WMMA GEMMs with async global->LDS
// double buffering + edge-parallel GCN scatter.
// ---------------------------------------------------------------------------

typedef __attribute__((ext_vector_type(2))) float v2f;
typedef __attribute__((ext_vector_type(8))) float v8f;
typedef int v4i_gcc __attribute__((vector_size(16)));  // matches builtin param

#define NNODES 50000
#define INC    3000
#define HID    256   // packed [W_rna1 | W_rna2] output width
#define OUTC   32

// ---- CDNA5 async global->LDS path (guarded; sync float4 fallback) ----------
#if defined(__has_builtin)
#if __has_builtin(__builtin_amdgcn_global_load_async_to_lds_b128)
#define HAS_ASYNC 1
#endif
#endif

#ifdef HAS_ASYNC
#if __has_builtin(__builtin_amdgcn_s_wait_asynccnt)
#define WAIT_ASYNC(n) __builtin_amdgcn_s_wait_asynccnt(n)
#else
#define WAIT_ASYNC(n) asm volatile("s_wait_asynccnt %0" ::"n"(n) : "memory")
#endif
#else
#define WAIT_ASYNC(n) ((void)0)
#endif

__device__ __forceinline__ void async_cp16(const float* __restrict__ g,
                                           float* __restrict__ l) {
#ifdef HAS_ASYNC
  __builtin_amdgcn_global_load_async_to_lds_b128(
      (__attribute__((address_space(1))) v4i_gcc*)g,
      (__attribute__((address_space(3))) v4i_gcc*)l, /*offset=*/0, /*cpol=*/0);
#else
  *(float4*)l = *(const float4*)g;
#endif
}

__device__ __forceinline__ v8f wmma4(v2f a, v2f b, v8f c) {
  // V_WMMA_F32_16X16X4_F32 : D = A(16x4) x B(4x16) + C(16x16), all fp32.
  return __builtin_amdgcn_wmma_f32_16x16x4_f32(
      /*neg_a=*/false, a, /*neg_b=*/false, b,
      /*c_mod=*/(short)0, c, /*reuse_a=*/false, /*reuse_b=*/false);
}

// ---------------------------------------------------------------------------
// Big GEMM: H[N,256] = X[N,3000] @ W[3000,256], weights pre-transposed to
// Wct[256][3000] so both A and B LDS tiles are K-contiguous: every WMMA
// fragment is one aligned ds_load_b64 (no VGPR shuffle to form operand pairs).
//
// Block = 256 threads = 8 waves, computes a 16-row x 256-col tile.
// K is staged in chunks of 40 through double-buffered LDS via async B128 DMA
// (ASYNCcnt), overlapping the next chunk's copy with this chunk's WMMAs.
// Each wave issues exactly 11 async copies per chunk (10 B + 1 masked A), so
// s_wait_asynccnt 11 after issuing chunk i+1 guarantees chunk i has landed.
// Row stride 44: rows 16B-aligned for the DMA; lane bank stride 44 (≡ distinct
// mod 64) -> conflict-free b64 reads; half-waves in disjoint mod-4 classes.
// ---------------------------------------------------------------------------
#define KC    40
#define AS_LD 44
#define BT_LD 44
#define NKCH  (INC / KC)  // 75

__global__ __launch_bounds__(256) void k_gemm_big(const float* __restrict__ X,
                                                  const float* __restrict__ Wct,
                                                  float* __restrict__ H) {
  __shared__ __align__(16) float As[2][16 * AS_LD];
  __shared__ __align__(16) float Bs[2][HID * BT_LD];
  const int row0    = blockIdx.x * 16;
  const int tid     = threadIdx.x;
  const int wave    = tid >> 5;
  const int lane    = tid & 31;
  const int m       = lane & 15;         // row (A) / col (B) within tile
  const int koff    = (lane >> 4) << 1;  // 0 or 2
  const int colbase = wave * 32;

  // Staging: B -> thread owns one column (40 contiguous K values = 10 chunks).
  const int bcol = tid;
  // A: 160 16B chunks spread as 20 per wave -> every wave issues exactly 1.
  const int aidx  = wave * 20 + lane;
  const int ar    = aidx / 10;
  const int ach   = aidx - ar * 10;
  const bool a_act = (lane < 20);

  auto stage = [&](int buf, int k0) {
    const float* bsrc = Wct + (size_t)bcol * INC + k0;
    float* bdst = &Bs[buf][bcol * BT_LD];
#pragma unroll
    for (int ch = 0; ch < 10; ++ch)
      async_cp16(bsrc + (ch << 2), bdst + (ch << 2));
    if (a_act)
      async_cp16(X + (size_t)(row0 + ar) * INC + k0 + (ach << 2),
                 &As[buf][ar * AS_LD + (ach << 2)]);
  };

  stage(0, 0);
  v8f c0 = {};
  v8f c1 = {};

  for (int i = 0; i < NKCH; ++i) {
    const int buf = i & 1;
    if (i + 1 < NKCH) {
      stage(buf ^ 1, (i + 1) * KC);
      WAIT_ASYNC(11);   // chunk i fully in LDS (chunk i+1 still in flight)
    } else {
      WAIT_ASYNC(0);
    }
    __syncthreads();

    const float* Ab = &As[buf][m * AS_LD + koff];
    const float* B0 = &Bs[buf][(colbase + m) * BT_LD + koff];
    const float* B1 = B0 + 16 * BT_LD;
#pragma unroll
    for (int kk = 0; kk < KC; kk += 4) {
      v2f a  = *(const v2f*)(Ab + kk);
      v2f b0 = *(const v2f*)(B0 + kk);
      v2f b1 = *(const v2f*)(B1 + kk);
      c0 = wmma4(a, b0, c0);
      c1 = wmma4(a, b1, c1);
    }
    __syncthreads();
  }

  const int rbase = row0 + ((lane >> 4) << 3);
#pragma unroll
  for (int i = 0; i < 8; ++i) {
    float* hp = H + (size_t)(rbase + i) * HID + colbase + m;
    hp[0]  = c0[i];
    hp[16] = c1[i];
  }
}

// ---------------------------------------------------------------------------
// Small GEMM: C[N,32] = A[N,K] @ B[K,32] (+bias). One wave per 16x16 tile.
// K must be a multiple of 4. grid = (N/16, 2).
// ---------------------------------------------------------------------------
__global__ __launch_bounds__(32) void k_gemm_small(const float* __restrict__ A,
                                                   const float* __restrict__ B,
                                                   const float* __restrict__ bias,
                                                   float* __restrict__ C, int K) {
  const int row0    = blockIdx.x * 16;
  const int colbase = blockIdx.y * 16;
  const int lane    = threadIdx.x;
  const int m       = lane & 15;
  const int koff    = (lane >> 4) << 1;
  v8f c = {};
  const float* ap = A + (size_t)(row0 + m) * K + koff;
  const float* bp = B + (size_t)koff * OUTC + colbase + m;
  for (int k = 0; k < K; k += 4) {
    v2f a, b;
    a.x = ap[k];
    a.y = ap[k + 1];
    b.x = bp[(size_t)k * OUTC];
    b.y = bp[(size_t)(k + 1) * OUTC];
    c = wmma4(a, b, c);
  }
  const int rbase = row0 + ((lane >> 4) << 3);
  const float bv  = bias ? bias[colbase + m] : 0.0f;
#pragma unroll
  for (int i = 0; i < 8; ++i)
    C[(size_t)(rbase + i) * OUTC + colbase + m] = c[i] + bv;
}

// ---------------------------------------------------------------------------
// Tiny dense projection (K = 30 or 50, not WMMA-friendly): thread per (i,col).
// ---------------------------------------------------------------------------
__global__ void k_dense_small(const float* __restrict__ X, int K,
                              const float* __restrict__ W,
                              float* __restrict__ Hout, int total) {
  int gid = blockIdx.x * blockDim.x + threadIdx.x;
  if (gid >= total) return;
  int i   = gid >> 5;
  int col = gid & 31;
  const float* xr = X + (size_t)i * K;
  float acc = 0.0f;
  for (int k = 0; k < K; ++k) acc += xr[k] * W[k * OUTC + col];
  Hout[gid] = acc;
}

// ---------------------------------------------------------------------------
// Elementwise / graph kernels
// ---------------------------------------------------------------------------
__global__ void k_zero(float* __restrict__ p, int n) {
  int i = blockIdx.x * blockDim.x + threadIdx.x;
  if (i < n) p[i] = 0.0f;
}

__global__ void k_deg(const int* __restrict__ dst, const float* __restrict__ ew,
                      float* __restrict__ deg, int E) {
  int e = blockIdx.x * blockDim.x + threadIdx.x;
  if (e < E) atomicAdd(&deg[dst[e]], ew[e]);
}

__global__ void k_rsqrt(float* __restrict__ d, int n) {
  int i = blockIdx.x * blockDim.x + threadIdx.x;
  if (i < n) d[i] = rsqrtf(d[i] + 1.0f);  // +1 = self-loop weight
}

// Wct[c][k] = (c<128 ? W_rna1 : W_rna2)[k][c%128]  -- packed AND transposed
__global__ void k_pack_t(const float* __restrict__ W1, const float* __restrict__ W2,
                         float* __restrict__ Wct, int total) {
  int idx = blockIdx.x * blockDim.x + threadIdx.x;
  if (idx >= total) return;
  int c = idx / INC;
  int k = idx - c * INC;
  Wct[idx] = (c < 128) ? W1[(size_t)k * 128 + c] : W2[(size_t)k * 128 + (c - 128)];
}

// out[i,f] = Hsrc[i, off+f] * dinv[i]^2   (self-loop contribution, non-atomic init)
__global__ void k_selfloop(const float* __restrict__ Hsrc, int ldH, int off,
                           const float* __restrict__ dinv, float* __restrict__ outp,
                           int F, int total) {
  int gid = blockIdx.x * blockDim.x + threadIdx.x;
  if (gid >= total) return;
  int i = gid / F;
  int f = gid - i * F;
  float dv = dinv[i];
  outp[gid] = Hsrc[(size_t)i * ldH + off + f] * dv * dv;
}

// acc[dst, f] += Hsrc[src, off+f] * ew * dinv[src] * dinv[dst]; float4 per thread
__global__ void k_scatter(const float* __restrict__ Hsrc, int ldH, int off,
                          const int* __restrict__ src, const int* __restrict__ dst,
                          const float* __restrict__ ew, const float* __restrict__ dinv,
                          float* __restrict__ acc, int ldA, int F4, int total) {
  int gid = blockIdx.x * blockDim.x + threadIdx.x;
  if (gid >= total) return;
  int e  = gid / F4;
  int f4 = gid - e * F4;
  int s = src[e], d = dst[e];
  float w = ew[e] * dinv[s] * dinv[d];
  const float4 h = *(const float4*)(Hsrc + (size_t)s * ldH + off + (f4 << 2));
  float* p = acc + (size_t)d * ldA + (f4 << 2);
  atomicAdd(p + 0, h.x * w);
  atomicAdd(p + 1, h.y * w);
  atomicAdd(p + 2, h.z * w);
  atomicAdd(p + 3, h.w * w);
}

__global__ void k_bias(float* __restrict__ p, const float* __restrict__ b,
                       int Fmask, int do_relu, int total) {
  int gid = blockIdx.x * blockDim.x + threadIdx.x;
  if (gid >= total) return;
  float v = p[gid] + b[gid & Fmask];
  p[gid] = do_relu ? fmaxf(v, 0.0f) : v;
}

// comb[i, 0:128] = [x_sim[i] | x_dist[i] | pro[i] | atac[i]]
__global__ void k_comb(const float* __restrict__ a, const float* __restrict__ b,
                       const float* __restrict__ c, const float* __restrict__ d,
                       float* __restrict__ comb, int total) {
  int gid = blockIdx.x * blockDim.x + threadIdx.x;
  if (gid >= total) return;
  int i = gid >> 7;
  int f = gid & 127;
  const float* s = (f < 64) ? ((f < 32) ? a : b) : ((f < 96) ? c : d);
  comb[gid] = s[(size_t)i * 32 + (f & 31)];
}

static inline int cdiv(long long a, int b) { return (int)((a + b - 1) / b); }

// ---------------------------------------------------------------------------
extern "C" void kernel_launch(void* const* d_in, const int* in_sizes, int n_in,
                              void* d_out, int out_size, void* d_ws, size_t ws_size,
                              hipStream_t stream) {
  const float* x_RNA  = (const float*)d_in[0];
  const float* x_ADT  = (const float*)d_in[1];
  const float* x_ATAC = (const float*)d_in[2];
  const int*   sim_ei  = (const int*)d_in[3];
  const float* sim_ew  = (const float*)d_in[4];
  const int*   dist_ei = (const int*)d_in[5];
  const float* dist_ew = (const float*)d_in[6];
  const int*   com_ei  = (const int*)d_in[7];
  const float* com_ew  = (const float*)d_in[8];
  const float* W1   = (const float*)d_in[9];  const float* b1    = (const float*)d_in[10];
  const float* W2   = (const float*)d_in[11]; const float* b2    = (const float*)d_in[12];
  const float* Wsim = (const float*)d_in[13]; const float* bsim  = (const float*)d_in[14];
  const float* Wdst = (const float*)d_in[15]; const float* bdst  = (const float*)d_in[16];
  const float* Wpro = (const float*)d_in[17]; const float* bpro  = (const float*)d_in[18];
  const float* Wata = (const float*)d_in[19]; const float* bata  = (const float*)d_in[20];
  const float* Wfus = (const float*)d_in[21]; const float* bfus  = (const float*)d_in[22];

  const int N     = NNODES;
  const int Esim  = in_sizes[4];
  const int Edist = in_sizes[6];
  const int Ecom  = in_sizes[8];
  const int* sim_s  = sim_ei;            const int* sim_d  = sim_ei + Esim;
  const int* dist_s = dist_ei;           const int* dist_d = dist_ei + Edist;
  const int* com_s  = com_ei;            const int* com_d  = com_ei + Ecom;

  float* out = (float*)d_out;
  float* o_xsim  = out;
  float* o_xdist = out + (size_t)N * 32;
  float* o_fused = out + (size_t)N * 32 * 2;
  float* o_pro   = out + (size_t)N * 32 * 3;
  float* o_atac  = out + (size_t)N * 32 * 4;

  float* ws = (float*)d_ws;
  size_t off = 0;
  float* dinv_sim  = ws + off; off += N;                 // deg -> dinv in place
  float* dinv_dist = ws + off; off += N;
  float* dinv_com  = ws + off; off += N;
  float* Wct   = ws + off; off += (size_t)HID * INC;     // 256 x 3000 (transposed)
  float* H     = ws + off; off += (size_t)N * HID;       // [N,256]
  float* XS    = ws + off; off += (size_t)N * 128;       // reused later as comb
  float* XD    = ws + off; off += (size_t)N * 128;
  float* HS2   = ws + off; off += (size_t)N * 32;
  float* HD2   = ws + off; off += (size_t)N * 32;
  float* Hpro  = ws + off; off += (size_t)N * 32;
  float* Hatac = ws + off; off += (size_t)N * 32;
  float* comb  = XS;  // XS is dead after HS2 is computed

  const int T = 256;

  // --- degrees + symmetric norm ---
  k_zero<<<cdiv(3 * N, T), T, 0, stream>>>(dinv_sim, 3 * N);
  k_deg<<<cdiv(Esim,  T), T, 0, stream>>>(sim_d,  sim_ew,  dinv_sim,  Esim);
  k_deg<<<cdiv(Edist, T), T, 0, stream>>>(dist_d, dist_ew, dinv_dist, Edist);
  k_deg<<<cdiv(Ecom,  T), T, 0, stream>>>(com_d,  com_ew,  dinv_com,  Ecom);
  k_rsqrt<<<cdiv(3 * N, T), T, 0, stream>>>(dinv_sim, 3 * N);

  // --- layer 1: one fused WMMA GEMM over packed+transposed weights ---
  k_pack_t<<<cdiv((long long)HID * INC, T), T, 0, stream>>>(W1, W2, Wct, HID * INC);
  k_gemm_big<<<N / 16, 256, 0, stream>>>(x_RNA, Wct, H);

  // self-loops, edge scatter, bias + ReLU
  k_selfloop<<<cdiv((long long)N * 128, T), T, 0, stream>>>(H, HID, 0,   dinv_sim,  XS, 128, N * 128);
  k_selfloop<<<cdiv((long long)N * 128, T), T, 0, stream>>>(H, HID, 128, dinv_dist, XD, 128, N * 128);
  k_scatter<<<cdiv((long long)Esim  * 32, T), T, 0, stream>>>(H, HID, 0,   sim_s,  sim_d,  sim_ew,  dinv_sim,  XS, 128, 32, Esim  * 32);
  k_scatter<<<cdiv((long long)Edist * 32, T), T, 0, stream>>>(H, HID, 128, dist_s, dist_d, dist_ew, dinv_dist, XD, 128, 32, Edist * 32);
  k_bias<<<cdiv((long long)N * 128, T), T, 0, stream>>>(XS, b1, 127, 1, N * 128);
  k_bias<<<cdiv((long long)N * 128, T), T, 0, stream>>>(XD, b2, 127, 1, N * 128);

  // --- layer 2: WMMA projections then aggregation ---
  dim3 gsm(N / 16, 2);
  k_gemm_small<<<gsm, 32, 0, stream>>>(XS, Wsim, nullptr, HS2, 128);
  k_gemm_small<<<gsm, 32, 0, stream>>>(XD, Wdst, nullptr, HD2, 128);
  k_dense_small<<<cdiv((long long)N * 32, T), T, 0, stream>>>(x_ADT,  30, Wpro, Hpro,  N * 32);
  k_dense_small<<<cdiv((long long)N * 32, T), T, 0, stream>>>(x_ATAC, 50, Wata, Hatac, N * 32);

  k_selfloop<<<cdiv((long long)N * 32, T), T, 0, stream>>>(HS2,   32, 0, dinv_sim,  o_xsim,  32, N * 32);
  k_selfloop<<<cdiv((long long)N * 32, T), T, 0, stream>>>(HD2,   32, 0, dinv_dist, o_xdist, 32, N * 32);
  k_selfloop<<<cdiv((long long)N * 32, T), T, 0, stream>>>(Hpro,  32, 0, dinv_com,  o_pro,   32, N * 32);
  k_selfloop<<<cdiv((long long)N * 32, T), T, 0, stream>>>(Hatac, 32, 0, dinv_com,  o_atac,  32, N * 32);
  k_scatter<<<cdiv((long long)Esim  * 8, T), T, 0, stream>>>(HS2,   32, 0, sim_s,  sim_d,  sim_ew,  dinv_sim,  o_xsim,  32, 8, Esim  * 8);
  k_scatter<<<cdiv((long long)Edist * 8, T), T, 0, stream>>>(HD2,   32, 0, dist_s, dist_d, dist_ew, dinv_dist, o_xdist, 32, 8, Edist * 8);
  k_scatter<<<cdiv((long long)Ecom  * 8, T), T, 0, stream>>>(Hpro,  32, 0, com_s,  com_d,  com_ew,  dinv_com,  o_pro,   32, 8, Ecom  * 8);
  k_scatter<<<cdiv((long long)Ecom  * 8, T), T, 0, stream>>>(Hatac, 32, 0, com_s,  com_d,  com_ew,  dinv_com,  o_atac,  32, 8, Ecom  * 8);
  k_bias<<<cdiv((long long)N * 32, T), T, 0, stream>>>(o_xsim,  bsim, 31, 0, N * 32);
  k_bias<<<cdiv((long long)N * 32, T), T, 0, stream>>>(o_xdist, bdst, 31, 0, N * 32);
  k_bias<<<cdiv((long long)N * 32, T), T, 0, stream>>>(o_pro,   bpro, 31, 0, N * 32);
  k_bias<<<cdiv((long long)N * 32, T), T, 0, stream>>>(o_atac,  bata, 31, 0, N * 32);

  // --- fuse ---
  k_comb<<<cdiv((long long)N * 128, T), T, 0, stream>>>(o_xsim, o_xdist, o_pro, o_atac, comb, N * 128);
  k_gemm_small<<<gsm, 32, 0, stream>>>(comb, Wfus, bfus, o_fused, 128);
}